// Sac4Dpl_22746146799707
// MI455X (gfx1250) — compile-verified
//
#include <hip/hip_runtime.h>
#include <math.h>

#define T_TOT  1825
#define NB     10000
#define WARM   365
#define T_OUT  (T_TOT - WARM)
#define CH     5

typedef float f2 __attribute__((ext_vector_type(2)));

struct Prm {
  float kc, pctim, adimp, uztwm, uzfwm, lztwm, lzfsm, lzfpm;
  float pfree, riva, zperc, rexp, uzk, lzsk, lzpk, ci, cgs, cgp;
  float inv_uztwm, inv_ul, inv_lztwm, inv_lzfpm, inv_lzfsm;
  float sum_uz, lzsum2, lzsum3, inv_lzsum3, ratio_p, parea;
  float pbi, pz;                  // pbase/uzfwm, pbase*zperc/uzfwm
  float opci, opcgs, opcgp;       // (1-ci)*parea etc.
  float c1, c2, c3;               // Muskingum coefficients
};

struct State {
  float auztw, alztw, uztw, uzfw, lztw, lzfs, lzfp;
  float qs, qi, qgs, qgp, mq;
  float isum;                     // qs+qi+qgs+qgp carried from previous step
};

template<bool STORE>
__device__ __forceinline__ void sac_step(const Prm& P, State& s,
                                         float p_in, float e_in,
                                         float& et_out, float& q_out)
{
  float p = fmaxf(p_in, 0.f);
  float e = isfinite(e_in) ? fmaxf(e_in, 0.f) : 0.f;   // nan_to_num + max(0)
  float ep = P.kc * e;

  // impervious/additional area water balance
  float ae1  = fminf(s.auztw, ep * (s.auztw * P.inv_uztwm));
  float ae3  = fmaxf((ep - ae1) * (s.alztw * P.inv_ul), 0.f);
  float t2   = (s.auztw - ae1) + p;                      // shared
  float pav  = fmaxf(t2 - P.uztwm, 0.f);
  float alz3 = s.alztw - ae3;
  float adsur = fmaxf(pav * (alz3 * P.inv_lztwm), 0.f);
  float t3   = pav - adsur + alz3;                       // shared
  float ars  = fmaxf(t3 - P.lztwm, 0.f);
  float auztw_n = fmaxf(fminf(P.uztwm, t2), 0.f);
  float alztw_n = fmaxf(fminf(P.lztwm, t3), 0.f);

  // evapotranspiration
  float e1  = fminf(s.uztw, ep * (s.uztw * P.inv_uztwm));
  float e2  = fmaxf(fminf(s.uzfw, ep - e1), 0.f);
  float e3  = fmaxf((ep - e1 - e2) * (s.lztw * P.inv_ul), 0.f);
  float lt1 = fmaxf(s.lztw - e3, 0.f);
  if (STORE) {
    float e4 = P.riva * ep;
    float ae2 = P.pctim * ep;
    et_out = ae2 + ae1 + ae3 + e1 + e2 + e3 + e4;
  }

  // upper-zone runoff
  float t4    = (s.uztw - e1) + p;                       // shared with ut
  float uzsum = t4 + (s.uzfw - e2);
  float rs = fmaxf(uzsum - P.sum_uz, 0.f) * P.parea;
  float ut = fmaxf(fminf(P.uztwm, t4), 0.f);
  float uf = fmaxf(fminf(P.uzfwm, uzsum - ut), 0.f);
  float ri = uf * P.uzk;
  uf = fmaxf(uf - ri, 0.f);

  // percolation: v_log_f32/v_exp_f32 pow (defr in [0,1], rexp in [1,4])
  // log2(0) = -inf -> exp2(-inf) = 0, matching jnp.power(0, r>0) = 0
  float lzsum = s.lzfs + s.lzfp + lt1;
  float defr  = fmaxf(1.f - lzsum * P.inv_lzsum3, 0.f);
  float pw    = __builtin_amdgcn_exp2f(P.rexp * __builtin_amdgcn_logf(defr));
  float perc  = (P.pbi + P.pz * pw) * uf;
  float rate  = fmaxf(fminf(perc, P.lzsum3 - lzsum), 0.f);
  uf = fmaxf(uf - rate, 0.f);

  float fx = fmaxf(fminf(P.lzsum2 - (s.lzfs + s.lzfp),
                         fmaxf(rate - (P.lztwm - lt1), rate * P.pfree)), 0.f);
  float perct = rate - fx;
  float a  = 1.f - s.lzfp * P.inv_lzfpm;
  float bb = 1.f - s.lzfs * P.inv_lzfsm;
  // v_rcp_f32; a+bb==0 -> inf, (a+a)*inf with a==0 -> NaN (matches jnp 0/0)
  float coef = P.ratio_p * ((a + a) * __builtin_amdgcn_rcpf(a + bb));
  coef = (coef > 1.f) ? 1.f : coef;                // NaN-propagating min
  float percp = fmaxf(fminf(P.lzfpm - s.lzfp,
                            fmaxf(fx - (P.lzfsm - s.lzfs), coef * fx)), 0.f);
  float percs = fmaxf(fx - percp, 0.f);

  float lt = fminf(lt1 + perct, P.lztwm);
  float ls = s.lzfs + percs;
  float lp = s.lzfp + percp;
  float rgs = ls * P.lzsk;  ls = fmaxf(ls - rgs, 0.f);
  float rgp = lp * P.lzpk;  lp = fmaxf(lp - rgp, 0.f);

  // routing (i1 == previous step's i2, carried in s.isum)
  float roimp = P.pctim * p;
  float qs_n  = roimp + (adsur + ars) * P.adimp + rs;
  float qi_n  = P.ci  * s.qi  + P.opci  * ri;
  float qgs_n = P.cgs * s.qgs + P.opcgs * rgs;
  float qgp_n = P.cgp * s.qgp + P.opcgp * rgp;
  float i2 = qs_n + qi_n + qgs_n + qgp_n;
  float o2 = P.c1 * s.isum + P.c2 * i2 + P.c3 * s.mq;
  q_out = o2;

  s.auztw = auztw_n; s.alztw = alztw_n;
  s.uztw = ut; s.uzfw = uf; s.lztw = lt; s.lzfs = ls; s.lzfp = lp;
  s.qs = qs_n; s.qi = qi_n; s.qgs = qgs_n; s.qgp = qgp_n;
  s.mq = o2; s.isum = i2;
}

// Load CH timesteps; u*NB*8B folds into the 24-bit instruction offset.
__device__ __forceinline__ void load_chunk(f2 (&buf)[CH], const f2* __restrict__ pe2,
                                           int t, int b)
{
  const f2* base = pe2 + (size_t)t * NB + b;
#pragma unroll
  for (int u = 0; u < CH; ++u)
    buf[u] = __builtin_nontemporal_load(base + (size_t)u * NB);
}

// Compute CH steps; uniform scalar output base `ob` -> saddr-form stores.
template<bool STORE>
__device__ __forceinline__ void comp_chunk(const f2 (&buf)[CH], const Prm& P, State& s,
                                           float* __restrict__ out, size_t& ob, int b)
{
#pragma unroll
  for (int u = 0; u < CH; ++u) {
    float et, q;
    sac_step<STORE>(P, s, buf[u].x, buf[u].y, et, q);
    if (STORE) {
      __builtin_nontemporal_store(q,  out + ob + (size_t)u * NB + b);
      __builtin_nontemporal_store(et, out + (size_t)T_OUT * NB + ob + (size_t)u * NB + b);
    }
  }
  if (STORE) ob += (size_t)CH * NB;
}

__global__ __launch_bounds__(64)
void sac_scan_kernel(const float* __restrict__ pe,       // (T, B, 2)
                     const float* __restrict__ params,   // (B, 21)
                     float* __restrict__ out)            // (2, T_OUT, B)
{
  const int b = blockIdx.x * blockDim.x + threadIdx.x;
  if (b >= NB) return;

  const float LO[21] = {0.1f,0.0f,0.0f,10.0f,10.0f,50.0f,10.0f,50.0f,0.0f,0.0f,0.0f,
                        5.0f,1.0f,0.1f,0.01f,0.001f,0.5f,0.95f,0.98f,0.0f,0.0f};
  const float HI[21] = {1.2f,0.1f,0.3f,100.0f,100.0f,400.0f,100.0f,1000.0f,0.3f,0.5f,0.1f,
                        350.0f,4.0f,0.5f,0.35f,0.05f,0.9f,0.998f,0.998f,1.0f,0.5f};
  float pr[21];
#pragma unroll
  for (int i = 0; i < 21; ++i)
    pr[i] = LO[i] + params[(size_t)b * 21 + i] * (HI[i] - LO[i]);

  Prm P;
  P.kc = pr[0];  P.pctim = pr[1];  P.adimp = pr[2];
  P.uztwm = pr[3]; P.uzfwm = pr[4]; P.lztwm = pr[5];
  P.lzfsm = pr[6]; P.lzfpm = pr[7];
  P.pfree = pr[9]; P.riva = pr[10];
  P.zperc = pr[11]; P.rexp = pr[12]; P.uzk = pr[13];
  P.lzsk = pr[14]; P.lzpk = pr[15];
  P.ci = pr[16]; P.cgs = pr[17]; P.cgp = pr[18];
  const float ke = pr[19], xe = pr[20];

  P.inv_uztwm = 1.f / P.uztwm;
  P.inv_ul    = 1.f / (P.uztwm + P.lztwm);
  P.inv_lztwm = 1.f / P.lztwm;
  P.inv_lzfpm = 1.f / P.lzfpm;
  P.inv_lzfsm = 1.f / P.lzfsm;
  P.sum_uz    = P.uztwm + P.uzfwm;
  P.lzsum2    = P.lzfsm + P.lzfpm;
  P.lzsum3    = P.lzfsm + P.lzfpm + P.lztwm;
  P.inv_lzsum3= 1.f / P.lzsum3;
  float pbase = P.lzfsm * P.lzsk + P.lzfpm * P.lzpk;
  float inv_uzfwm = 1.f / P.uzfwm;
  P.pbi = pbase * inv_uzfwm;
  P.pz  = pbase * P.zperc * inv_uzfwm;
  P.ratio_p = P.lzfpm / P.lzsum2;
  P.parea   = 1.f - P.pctim - P.adimp;
  P.opci  = (1.f - P.ci)  * P.parea;
  P.opcgs = (1.f - P.cgs) * P.parea;
  P.opcgp = (1.f - P.cgp) * P.parea;
  const float dt = 0.5f;
  const float denom = ke * (1.f - xe) + dt;
  P.c1 = (ke * xe + dt) / denom;
  P.c2 = (dt - ke * xe) / denom;
  P.c3 = (ke * (1.f - xe) - dt) / denom;

  State s;
  s.auztw = s.alztw = s.uztw = s.uzfw = s.lztw = s.lzfs = s.lzfp = 0.01f;
  s.qs = s.qi = s.qgs = s.qgp = s.mq = 0.01f;
  s.isum = 0.04f;                      // qs+qi+qgs+qgp at t=0

  const f2* __restrict__ pe2 = (const f2*)pe;
  size_t ob = 0;                        // uniform output base (floats)

  f2 A[CH], B[CH];
  int t = 0;

  // ---- warmup: chunks 0..355 (72 chunks, no stores), double buffered ----
  load_chunk(A, pe2, 0, b);
#pragma unroll 1
  for (int i = 0; i < 36; ++i) {
    load_chunk(B, pe2, t + CH, b);
    comp_chunk<false>(A, P, s, out, ob, b);
    load_chunk(A, pe2, t + 2 * CH, b);
    comp_chunk<false>(B, P, s, out, ob, b);
    t += 2 * CH;
  }
  // t == 360, A holds chunk 360 (last warmup chunk)
  load_chunk(B, pe2, 365, b);
  comp_chunk<false>(A, P, s, out, ob, b);   // 360..364 (warmup done)
  load_chunk(A, pe2, 370, b);
  comp_chunk<true>(B, P, s, out, ob, b);    // 365..369 stored
  t = 370;

  // ---- output: chunks 370..1815 (290 chunks), double buffered ----
#pragma unroll 1
  for (int i = 0; i < 145; ++i) {
    load_chunk(B, pe2, t + CH, b);
    comp_chunk<true>(A, P, s, out, ob, b);
    load_chunk(A, pe2, t + 2 * CH, b);
    comp_chunk<true>(B, P, s, out, ob, b);
    t += 2 * CH;
  }
  // t == 1820, A holds final chunk 1820..1824
  comp_chunk<true>(A, P, s, out, ob, b);
}

extern "C" void kernel_launch(void* const* d_in, const int* in_sizes, int n_in,
                              void* d_out, int out_size, void* d_ws, size_t ws_size,
                              hipStream_t stream) {
  const float* pe     = (const float*)d_in[0];   // (1825, 10000, 2) f32
  const float* params = (const float*)d_in[1];   // (10000, 21) f32
  float* out = (float*)d_out;                    // (2, 1460, 10000) f32

  const int block = 64;
  const int grid  = (NB + block - 1) / block;    // 157 blocks
  sac_scan_kernel<<<grid, block, 0, stream>>>(pe, params, out);
}